// NormQKNorm_317827580176
// MI455X (gfx1250) — compile-verified
//
#include <hip/hip_runtime.h>
#include <math.h>

// MI455X / gfx1250: wave32, WMMA f32 16x16x4 (fp32 matrix path — workload is
// HBM-bandwidth-bound at ~185MB / 1.5GFLOP, so fp32 WMMA is the right ceiling).

typedef __attribute__((ext_vector_type(2))) float v2f;
typedef __attribute__((ext_vector_type(8))) float v8f;

#define MM 16
#define NN 2048
#define DD 128
#define PP 8192
#define HH 16
#define NCHUNK 33   // 32 cache chunks of 256 rows + 1 chunk of 16 new rows

// ---------------------------------------------------------------------------
// Kernel 1: RMSNorm of X rows (16 rows of 2048)
// ---------------------------------------------------------------------------
__global__ __launch_bounds__(256) void k_rmsnorm_x(const float* __restrict__ X,
                                                   float* __restrict__ Xn)
{
    __shared__ float red[256];
    const int m = blockIdx.x;
    const int t = threadIdx.x;
    float ss = 0.0f;
    for (int j = t; j < NN; j += 256) {
        float x = X[(size_t)m * NN + j];
        ss += x * x;
    }
    red[t] = ss;
    __syncthreads();
    for (int o = 128; o > 0; o >>= 1) {
        if (t < o) red[t] += red[t + o];
        __syncthreads();
    }
    const float rs = rsqrtf(red[0] / (float)NN);
    for (int j = t; j < NN; j += 256)
        Xn[(size_t)m * NN + j] = X[(size_t)m * NN + j] * rs;
}

// ---------------------------------------------------------------------------
// Kernel 2: projections Q/K/V = Xn @ W via V_WMMA_F32_16X16X4_F32.
// One wave per 16x16 output tile; 384 tiles total; each W element read once.
// A frag: lane L holds Xn[L%16][kb + 2*(L/16) + r]  (contiguous float2)
// B frag: lane L holds W[kb + 2*(L/16) + r][n0 + L%16]
// D:      lane L reg r -> out[r + 8*(L/16)][n0 + L%16]
// ---------------------------------------------------------------------------
__global__ __launch_bounds__(32) void k_proj_wmma(const float* __restrict__ Xn,
                                                  const float* __restrict__ Wq,
                                                  const float* __restrict__ Wk,
                                                  const float* __restrict__ Wv,
                                                  float* __restrict__ qraw,
                                                  float* __restrict__ kraw,
                                                  float* __restrict__ vraw)
{
    const int lane = threadIdx.x;
    const int col  = lane & 15;
    const int lh   = lane >> 4;
    const int bid  = blockIdx.x;
    const int w    = bid >> 7;           // 0=Q, 1=K, 2=V
    const int n0   = (bid & 127) << 4;   // output column tile

    const float* W   = (w == 0) ? Wq : (w == 1) ? Wk : Wv;
    float*       out = (w == 0) ? qraw : (w == 1) ? kraw : vraw;

    const float* arow = Xn + (size_t)col * NN + 2 * lh;
    const float* bcol = W + n0 + col;

    v8f acc = {};
    #pragma unroll 8
    for (int kb = 0; kb < NN; kb += 4) {
        float2 a2 = *reinterpret_cast<const float2*>(arow + kb);
        v2f A = {a2.x, a2.y};
        const float* bp = bcol + (size_t)(kb + 2 * lh) * NN;
        v2f B = {bp[0], bp[NN]};
        acc = __builtin_amdgcn_wmma_f32_16x16x4_f32(false, A, false, B,
                                                    (short)0, acc, false, false);
    }

    #pragma unroll
    for (int r = 0; r < 8; ++r)
        out[(size_t)(r + 8 * lh) * NN + n0 + col] = acc[r];
}

// ---------------------------------------------------------------------------
// Kernel 3: QK-RMSNorm over D=128 and re-layout to [H][M][D]; V re-layout only.
// One wave (32 lanes) per (h,m); each lane owns 4 consecutive d.
// ---------------------------------------------------------------------------
__global__ __launch_bounds__(256) void k_qknorm(const float* __restrict__ qraw,
                                                const float* __restrict__ kraw,
                                                const float* __restrict__ vraw,
                                                float* __restrict__ qn,
                                                float* __restrict__ kn,
                                                float* __restrict__ vn)
{
    const int gtid = blockIdx.x * blockDim.x + threadIdx.x;
    const int wid  = gtid >> 5;   // 0..255 => (h,m)
    const int lane = gtid & 31;
    const int h = wid >> 4;
    const int m = wid & 15;
    const int d0 = lane * 4;

    const size_t src = (size_t)m * NN + (size_t)h * DD + d0;
    const size_t dst = ((size_t)h * MM + m) * DD + d0;

    // Q
    float4 q4 = *reinterpret_cast<const float4*>(qraw + src);
    float ss = q4.x * q4.x + q4.y * q4.y + q4.z * q4.z + q4.w * q4.w;
    #pragma unroll
    for (int mk = 16; mk >= 1; mk >>= 1) ss += __shfl_xor(ss, mk, 32);
    float rs = rsqrtf(ss / (float)DD);
    float4 o;
    o.x = q4.x * rs; o.y = q4.y * rs; o.z = q4.z * rs; o.w = q4.w * rs;
    *reinterpret_cast<float4*>(qn + dst) = o;

    // K
    float4 k4 = *reinterpret_cast<const float4*>(kraw + src);
    ss = k4.x * k4.x + k4.y * k4.y + k4.z * k4.z + k4.w * k4.w;
    #pragma unroll
    for (int mk = 16; mk >= 1; mk >>= 1) ss += __shfl_xor(ss, mk, 32);
    rs = rsqrtf(ss / (float)DD);
    o.x = k4.x * rs; o.y = k4.y * rs; o.z = k4.z * rs; o.w = k4.w * rs;
    *reinterpret_cast<float4*>(kn + dst) = o;

    // V: straight re-layout
    *reinterpret_cast<float4*>(vn + dst) =
        *reinterpret_cast<const float4*>(vraw + src);
}

// ---------------------------------------------------------------------------
// Kernel 4: split-K attention partials. grid = (33 chunks, 16 heads), 1 wave.
// Per block: scores[16, nrows] via WMMA into LDS, per-row max/sumexp,
// exp-probs back in LDS, then partial out[16,128] = probs @ V via WMMA.
// ---------------------------------------------------------------------------
__global__ __launch_bounds__(32) void k_attn_partial(const float* __restrict__ qn,
                                                     const float* __restrict__ kn,
                                                     const float* __restrict__ vn,
                                                     const float* __restrict__ cK,
                                                     const float* __restrict__ cV,
                                                     float* __restrict__ pout,
                                                     float* __restrict__ pmax,
                                                     float* __restrict__ psum)
{
    __shared__ __align__(16) float q_lds[MM * DD];   // 16 x 128  (8 KB)
    __shared__ __align__(16) float sc[MM * 256];     // 16 x 256  (16 KB)

    const int lane = threadIdx.x;
    const int col  = lane & 15;
    const int lh   = lane >> 4;
    const int s    = blockIdx.x;  // chunk
    const int h    = blockIdx.y;  // head

    // stage q tile for this head into LDS
    const float* qh = qn + (size_t)h * MM * DD;
    for (int i = lane; i < (MM * DD) / 4; i += 32)
        reinterpret_cast<float4*>(q_lds)[i] =
            reinterpret_cast<const float4*>(qh)[i];

    const int nrows = (s < 32) ? 256 : MM;
    const float* Kb = (s < 32) ? (cK + ((size_t)h * PP + (size_t)s * 256) * DD)
                               : (kn + (size_t)h * MM * DD);
    const float* Vb = (s < 32) ? (cV + ((size_t)h * PP + (size_t)s * 256) * DD)
                               : (vn + (size_t)h * MM * DD);

    if (nrows < 256) {
        for (int i = lane; i < MM * 256; i += 32) sc[i] = -3.0e38f;
    }

    // ---- Phase A: score tiles (q[16,128] @ K_rows[16,128]^T) ----
    for (int t = 0; t < nrows / 16; ++t) {
        const int j0 = t * 16;
        const float* krow = Kb + (size_t)(j0 + col) * DD + 2 * lh;
        const float* qrow = q_lds + (size_t)col * DD + 2 * lh;
        v8f acc = {};
        #pragma unroll
        for (int kb = 0; kb < DD; kb += 4) {
            float2 a2 = *reinterpret_cast<const float2*>(qrow + kb);
            float2 b2 = *reinterpret_cast<const float2*>(krow + kb);
            v2f A = {a2.x, a2.y};
            v2f B = {b2.x, b2.y};
            acc = __builtin_amdgcn_wmma_f32_16x16x4_f32(false, A, false, B,
                                                        (short)0, acc, false, false);
        }
        #pragma unroll
        for (int r = 0; r < 8; ++r)
            sc[(r + 8 * lh) * 256 + j0 + col] = acc[r];
    }

    // ---- per-row max / sumexp; probs (unnormalized exp) back into LDS ----
    // lane handles row=col, column half lh*128..lh*128+127
    float* scrow = sc + (size_t)col * 256 + lh * 128;
    float mloc = -3.0e38f;
    for (int j = 0; j < 128; ++j) mloc = fmaxf(mloc, scrow[j]);
    mloc = fmaxf(mloc, __shfl_xor(mloc, 16, 32));
    float ssum = 0.0f;
    for (int j = 0; j < 128; ++j) {
        float e = __expf(scrow[j] - mloc);
        scrow[j] = e;
        ssum += e;
    }
    ssum += __shfl_xor(ssum, 16, 32);
    if (lh == 0) {
        pmax[((size_t)h * NCHUNK + s) * MM + col] = mloc;
        psum[((size_t)h * NCHUNK + s) * MM + col] = ssum;
    }

    // ---- Phase B: partial_out[16,128] = probs[16,nrows] @ V[nrows,128] ----
    v8f zero = {};
    v8f oacc[8];
    #pragma unroll
    for (int dt = 0; dt < 8; ++dt) oacc[dt] = zero;

    for (int kb = 0; kb < nrows; kb += 4) {
        float2 a2 = *reinterpret_cast<const float2*>(sc + (size_t)col * 256 + kb + 2 * lh);
        v2f A = {a2.x, a2.y};
        const float* vrow = Vb + (size_t)(kb + 2 * lh) * DD + col;
        #pragma unroll
        for (int dt = 0; dt < 8; ++dt) {
            v2f B = {vrow[dt * 16], vrow[DD + dt * 16]};
            oacc[dt] = __builtin_amdgcn_wmma_f32_16x16x4_f32(false, A, false, B,
                                                             (short)0, oacc[dt],
                                                             false, false);
        }
    }

    float* po = pout + ((size_t)h * NCHUNK + s) * MM * DD;
    #pragma unroll
    for (int dt = 0; dt < 8; ++dt) {
        #pragma unroll
        for (int r = 0; r < 8; ++r)
            po[(size_t)(r + 8 * lh) * DD + dt * 16 + col] = oacc[dt][r];
    }
}

// ---------------------------------------------------------------------------
// Kernel 5: merge the 33 split-K partials per head with max-rescaling.
// ---------------------------------------------------------------------------
__global__ __launch_bounds__(256) void k_combine(const float* __restrict__ pout,
                                                 const float* __restrict__ pmax,
                                                 const float* __restrict__ psum,
                                                 float* __restrict__ out)
{
    const int h = blockIdx.x;
    for (int idx = threadIdx.x; idx < MM * DD; idx += blockDim.x) {
        const int m = idx >> 7;
        const int d = idx & (DD - 1);
        const float* pm  = pmax + (size_t)h * NCHUNK * MM + m;
        const float* psu = psum + (size_t)h * NCHUNK * MM + m;
        float g = -3.0e38f;
        for (int s = 0; s < NCHUNK; ++s) g = fmaxf(g, pm[(size_t)s * MM]);
        float tot = 0.0f, acc = 0.0f;
        for (int s = 0; s < NCHUNK; ++s) {
            float w = __expf(pm[(size_t)s * MM] - g);
            tot += w * psu[(size_t)s * MM];
            acc += w * pout[(((size_t)h * NCHUNK + s) * MM + m) * DD + d];
        }
        out[(size_t)m * NN + (size_t)h * DD + d] = acc / tot;
    }
}

// ---------------------------------------------------------------------------
extern "C" void kernel_launch(void* const* d_in, const int* in_sizes, int n_in,
                              void* d_out, int out_size, void* d_ws, size_t ws_size,
                              hipStream_t stream)
{
    (void)in_sizes; (void)n_in; (void)out_size; (void)ws_size;
    const float* X  = (const float*)d_in[0];
    const float* Wq = (const float*)d_in[1];
    const float* Wk = (const float*)d_in[2];
    const float* Wv = (const float*)d_in[3];
    const float* cK = (const float*)d_in[4];
    const float* cV = (const float*)d_in[5];
    float* out = (float*)d_out;
    float* ws  = (float*)d_ws;

    // workspace layout (floats); total ~5.3 MB
    float* xnorm = ws;                            // 16*2048
    float* qraw  = xnorm + MM * NN;               // 16*2048
    float* kraw  = qraw + MM * NN;
    float* vraw  = kraw + MM * NN;
    float* qn    = vraw + MM * NN;                // [H][M][D]
    float* kn    = qn + HH * MM * DD;
    float* vn    = kn + HH * MM * DD;
    float* pout  = vn + HH * MM * DD;             // [H][33][M][D]
    float* pmaxb = pout + (size_t)HH * NCHUNK * MM * DD;
    float* psumb = pmaxb + (size_t)HH * NCHUNK * MM;

    k_rmsnorm_x<<<16, 256, 0, stream>>>(X, xnorm);
    k_proj_wmma<<<3 * (NN / 16), 32, 0, stream>>>(xnorm, Wq, Wk, Wv,
                                                  qraw, kraw, vraw);
    k_qknorm<<<(HH * MM) / 8, 256, 0, stream>>>(qraw, kraw, vraw, qn, kn, vn);
    k_attn_partial<<<dim3(NCHUNK, HH), 32, 0, stream>>>(qn, kn, vn, cK, cV,
                                                        pout, pmaxb, psumb);
    k_combine<<<HH, 256, 0, stream>>>(pout, pmaxb, psumb, out);
}